// PointConvModule_26688926777501
// MI455X (gfx1250) — compile-verified
//
#include <hip/hip_runtime.h>
#include <cstdint>
#include <cstddef>

// ---------------------------------------------------------------------------
// CDNA5 (gfx1250) PointNet++-style pipeline.
// Matrix work runs on V_WMMA_F32_16X16X4_F32 (f32 in/out, 16x16 tiles, K step 4).
// Weights are pre-padded so the WMMA K-loop is branch-free (A = global_load_b64,
// B = 2x ds_load_b32, 1x v_wmma per step, unrolled x2).
// FPS is LDS-resident (320KB WGP LDS). wave32 everywhere.
// ---------------------------------------------------------------------------

typedef float v2f __attribute__((ext_vector_type(2)));
typedef float v8f __attribute__((ext_vector_type(8)));

__device__ __forceinline__ v8f wmma_f32_step(v2f a, v2f b, v8f c) {
  // 8-arg pattern: (neg_a, A, neg_b, B, c_mod, C, reuse_a, reuse_b)
  return __builtin_amdgcn_wmma_f32_16x16x4_f32(false, a, false, b, (short)0, c,
                                               false, false);
}

// ---------------------------------------------------------------------------
// BN fusion: scale = gamma / sqrt(var+eps), bias = beta - mean*scale
// ---------------------------------------------------------------------------
__global__ void bn_fuse_kernel(const float* __restrict__ gamma,
                               const float* __restrict__ beta,
                               const float* __restrict__ mean,
                               const float* __restrict__ var, int C,
                               float* __restrict__ scale,
                               float* __restrict__ bias) {
  int c = blockIdx.x * blockDim.x + threadIdx.x;
  if (c < C) {
    float inv = gamma[c] / sqrtf(var[c] + 1e-5f);
    scale[c] = inv;
    bias[c] = beta[c] - mean[c] * inv;
  }
}

// Zero-pad a (cout x cin) weight matrix to row stride cinP.
__global__ void pad_w_kernel(const float* __restrict__ W, int cout, int cin,
                             int cinP, float* __restrict__ Wp) {
  int t = blockIdx.x * blockDim.x + threadIdx.x;
  if (t < cout * cinP) {
    int m = t / cinP, k = t % cinP;
    Wp[t] = (k < cin) ? W[m * cin + k] : 0.0f;
  }
}

// ---------------------------------------------------------------------------
// Farthest point sampling: one workgroup per batch, everything in LDS.
// dynamic smem: px[N],py[N],pz[N],dist[N], rv[T], ri[T]
// ---------------------------------------------------------------------------
__global__ void fps_kernel(const float* __restrict__ xyz, int N, int npoint,
                           int* __restrict__ out) {
  extern __shared__ float smem[];
  const int T = blockDim.x;
  float* px = smem;
  float* py = px + N;
  float* pz = py + N;
  float* dist = pz + N;
  float* rv = dist + N;
  int* ri = (int*)(rv + T);
  __shared__ int lastS;

  const int b = blockIdx.x;
  const int tid = threadIdx.x;
  for (int i = tid; i < N; i += T) {
    px[i] = xyz[((long)b * N + i) * 3 + 0];
    py[i] = xyz[((long)b * N + i) * 3 + 1];
    pz[i] = xyz[((long)b * N + i) * 3 + 2];
    dist[i] = 1e10f;
  }
  if (tid == 0) lastS = 0;
  __syncthreads();

  int last = 0;
  for (int it = 0; it < npoint; ++it) {
    if (tid == 0) out[(long)b * npoint + it] = last;
    const float lx = px[last], ly = py[last], lz = pz[last];
    float best = -1.0f;
    int bi = 0;
    for (int i = tid; i < N; i += T) {
      float dx = px[i] - lx, dy = py[i] - ly, dz = pz[i] - lz;
      float d = dx * dx + dy * dy + dz * dz;
      float dm = fminf(dist[i], d);
      dist[i] = dm;
      if (dm > best) { best = dm; bi = i; }
    }
    rv[tid] = best;
    ri[tid] = bi;
    __syncthreads();
    for (int off = T >> 1; off > 0; off >>= 1) {
      if (tid < off) {
        float ov = rv[tid + off];
        int oi = ri[tid + off];
        if (ov > rv[tid] || (ov == rv[tid] && oi < ri[tid])) {
          rv[tid] = ov;
          ri[tid] = oi;
        }
      }
      __syncthreads();
    }
    if (tid == 0) lastS = ri[0];
    __syncthreads();
    last = lastS;
  }
}

// ---------------------------------------------------------------------------
// Gather sampled centers
// ---------------------------------------------------------------------------
__global__ void gather_xyz_kernel(const float* __restrict__ xyz,
                                  const int* __restrict__ inds, int Bn, int N,
                                  int S, float* __restrict__ out) {
  int t = blockIdx.x * blockDim.x + threadIdx.x;
  if (t < Bn * S) {
    int b = t / S, s = t % S;
    int i = inds[b * S + s];
    out[(long)t * 3 + 0] = xyz[((long)b * N + i) * 3 + 0];
    out[(long)t * 3 + 1] = xyz[((long)b * N + i) * 3 + 1];
    out[(long)t * 3 + 2] = xyz[((long)b * N + i) * 3 + 2];
  }
}

// ---------------------------------------------------------------------------
// Ball query: one wave32 per query point; first-64-in-index-order semantics
// reproduced via ballot + prefix-popcount.
// ---------------------------------------------------------------------------
__global__ void ball_query_kernel(const float* __restrict__ xyz,
                                  const float* __restrict__ new_xyz, int Bn,
                                  int N, int S, float r2,
                                  int* __restrict__ idxOut) {
  const int gw = (blockIdx.x * blockDim.x + threadIdx.x) >> 5;
  const int lane = threadIdx.x & 31;
  if (gw >= Bn * S) return;
  const int b = gw / S, s = gw % S;
  const float cx = new_xyz[((long)b * S + s) * 3 + 0];
  const float cy = new_xyz[((long)b * S + s) * 3 + 1];
  const float cz = new_xyz[((long)b * S + s) * 3 + 2];
  int* out = idxOut + ((long)b * S + s) * 64;

  int count = 0;
  int firstIdx = 0;
  for (int base = 0; base < N && count < 64; base += 32) {
    const int i = base + lane;  // N is a multiple of 32
    float dx = xyz[((long)b * N + i) * 3 + 0] - cx;
    float dy = xyz[((long)b * N + i) * 3 + 1] - cy;
    float dz = xyz[((long)b * N + i) * 3 + 2] - cz;
    bool inb = (dx * dx + dy * dy + dz * dz) < r2;
    unsigned mask = (unsigned)__ballot(inb);
    if (count == 0 && mask != 0u) firstIdx = base + (__ffs(mask) - 1);
    int prefix = __popc(mask & ((1u << lane) - 1u));
    int pos = count + prefix;
    if (inb && pos < 64) out[pos] = i;
    count += __popc(mask);
  }
  if (count > 64) count = 64;
  for (int j = count + lane; j < 64; j += 32) out[j] = firstIdx;
}

// ---------------------------------------------------------------------------
// LDS-to-LDS GEMM layer via V_WMMA_F32_16X16X4_F32, branch-free K loop.
//   Yout[m][n64] = relu(scale[m] * sum_k W[m][k]*Xin[k][n64] + bias[m])
// Xin is [kDimP][64] in LDS (kDimP % 4 == 0, padded rows are zero).
// W has row stride kDimP (pre-padded). cout % 16 == 0.
// A operand: one global_load_b64 per step (contiguous pair per lane).
// ---------------------------------------------------------------------------
__device__ __forceinline__ void gemm_relu_lds(
    const float* __restrict__ Xin, int kDimP, const float* __restrict__ W,
    const float* __restrict__ scale, const float* __restrict__ bias, int cout,
    float* __restrict__ Yout) {
  const int tid = threadIdx.x;
  const int wave = tid >> 5;
  const int lane = tid & 31;
  const int half = lane >> 4;
  const int l = lane & 15;
  const int nWaves = blockDim.x >> 5;
  const int nTilesTotal = (cout >> 4) * 4;

  for (int t = wave; t < nTilesTotal; t += nWaves) {
    const int m0 = (t >> 2) << 4;
    const int n0 = (t & 3) << 4;
    v8f acc;
#pragma unroll
    for (int r = 0; r < 8; ++r) acc[r] = 0.0f;

    // lane's A row, offset to its half's contiguous K pair
    const float* wr = W + (long)(m0 + l) * kDimP + half * 2;
    const float* xc = Xin + n0 + l + (half * 2) * 64;
    __builtin_prefetch(wr, 0, 0);  // global_prefetch_b8
#pragma unroll 2
    for (int k0 = 0; k0 < kDimP; k0 += 4) {
      v2f a = *(const v2f*)(wr + k0);  // global_load_b64
      v2f bm;
      bm.x = xc[k0 * 64];        // ds_load_b32
      bm.y = xc[k0 * 64 + 64];   // ds_load_b32
      acc = wmma_f32_step(a, bm, acc);
    }
#pragma unroll
    for (int r = 0; r < 8; ++r) {
      const int m = m0 + r + half * 8;
      float y = acc[r] * scale[m] + bias[m];
      Yout[m * 64 + n0 + l] = fmaxf(y, 0.0f);
    }
  }
}

// ---------------------------------------------------------------------------
// Fused SA stage: gather group -> 3 MLP layers (WMMA) -> maxpool over K=64.
// One workgroup (256 thr = 8 waves) per (batch, group).
// dynamic smem: X[cinP*64] | Y[cmax*64] | Z[cmax*64]
// W1 has row stride cinP (pre-padded); W2/W3 strides c1/c2.
// ---------------------------------------------------------------------------
__global__ void sa_fused_kernel(
    const float* __restrict__ xyz,      // B,N,3
    const float* __restrict__ feats,    // B,Cf,N (may be null when Cf==0)
    const float* __restrict__ new_xyz,  // B,S,3
    const int* __restrict__ ballIdx,    // B,S,64
    const float* __restrict__ W1, const float* __restrict__ s1,
    const float* __restrict__ b1, int c1, const float* __restrict__ W2,
    const float* __restrict__ s2, const float* __restrict__ b2, int c2,
    const float* __restrict__ W3, const float* __restrict__ s3,
    const float* __restrict__ b3, int c3, int Cf, int N, int S,
    float invRadius, float* __restrict__ outFeat /* B,c3,S */) {
  extern __shared__ float smem[];
  const int cin = 3 + Cf;
  const int cinP = (cin + 3) & ~3;
  int cmax = c1 > c2 ? c1 : c2;
  cmax = cmax > c3 ? cmax : c3;
  float* X = smem;
  float* Y = X + cinP * 64;
  float* Z = Y + cmax * 64;

  const int b = blockIdx.x / S;
  const int s = blockIdx.x % S;
  const int tid = threadIdx.x;

  const float cx = new_xyz[((long)b * S + s) * 3 + 0];
  const float cy = new_xyz[((long)b * S + s) * 3 + 1];
  const float cz = new_xyz[((long)b * S + s) * 3 + 2];
  const int* gidx = ballIdx + ((long)b * S + s) * 64;

  for (int j = tid; j < cinP * 64; j += blockDim.x) {
    const int c = j >> 6, k = j & 63;
    float v = 0.0f;
    if (c < 3) {
      const int i = gidx[k];
      const float ctr = (c == 0) ? cx : (c == 1) ? cy : cz;
      v = (xyz[((long)b * N + i) * 3 + c] - ctr) * invRadius;
    } else if (c < cin) {
      const int i = gidx[k];
      v = feats[((long)b * Cf + (c - 3)) * N + i];
    }
    X[c * 64 + k] = v;
  }
  __syncthreads();

  gemm_relu_lds(X, cinP, W1, s1, b1, c1, Y);
  __syncthreads();
  gemm_relu_lds(Y, c1, W2, s2, b2, c2, Z);
  __syncthreads();
  gemm_relu_lds(Z, c2, W3, s3, b3, c3, Y);
  __syncthreads();

  for (int c = tid; c < c3; c += blockDim.x) {
    float m = -1e30f;
#pragma unroll 8
    for (int k = 0; k < 64; ++k) m = fmaxf(m, Y[c * 64 + k]);
    outFeat[((long)b * c3 + c) * S + s] = m;
  }
}

// ---------------------------------------------------------------------------
// Pointwise MLP layer as a WMMA-tiled GEMM over points (Cin % 4 == 0):
//   Y[b][m][n] = relu(scale[m]*(sum_k W[m][k]*X[b][k][n]) + bias[m])
// Each wave owns one 16(cout) x 16(points) tile; branch-free K loop.
// ---------------------------------------------------------------------------
__global__ void pw_gemm_kernel(const float* __restrict__ Xin,  // B,Cin,Np
                               const float* __restrict__ W,    // Cout,Cin
                               const float* __restrict__ scale,
                               const float* __restrict__ bias,
                               float* __restrict__ Yout,  // B,Cout,Np
                               int Bn, int Cin, int Cout, int Np) {
  const int wave = threadIdx.x >> 5;
  const int lane = threadIdx.x & 31;
  const int half = lane >> 4;
  const int l = lane & 15;
  const long nTiles = Np >> 4;
  const long tilesPerBatch = (long)(Cout >> 4) * nTiles;
  const long total = (long)Bn * tilesPerBatch;
  long t = (long)blockIdx.x * (blockDim.x >> 5) + wave;
  if (t >= total) return;  // wave-uniform exit

  const int b = (int)(t / tilesPerBatch);
  const long r = t % tilesPerBatch;
  const int m0 = (int)(r / nTiles) << 4;
  const int n0 = (int)(r % nTiles) << 4;
  const float* Xb = Xin + (long)b * Cin * Np + n0 + l;
  const float* wr = W + (long)(m0 + l) * Cin + half * 2;
  __builtin_prefetch(wr, 0, 0);  // global_prefetch_b8

  v8f acc;
#pragma unroll
  for (int i = 0; i < 8; ++i) acc[i] = 0.0f;

#pragma unroll 2
  for (int k0 = 0; k0 < Cin; k0 += 4) {
    const long kA0 = k0 + half * 2;
    v2f a = *(const v2f*)(wr + k0);  // global_load_b64
    v2f bm;
    bm.x = Xb[kA0 * Np];
    bm.y = Xb[(kA0 + 1) * Np];
    acc = wmma_f32_step(a, bm, acc);
  }
#pragma unroll
  for (int rr = 0; rr < 8; ++rr) {
    const int m = m0 + rr + half * 8;
    float y = acc[rr] * scale[m] + bias[m];
    Yout[((long)b * Cout + m) * Np + n0 + l] = fmaxf(y, 0.0f);
  }
}

// ---------------------------------------------------------------------------
// 3-NN search (thread per unknown point) + interpolation weights
// ---------------------------------------------------------------------------
__global__ void nn3_kernel(const float* __restrict__ ux,
                           const float* __restrict__ kx, int Bn, int Nu,
                           int Nk, int* __restrict__ idx3,
                           float* __restrict__ w3) {
  long t = (long)blockIdx.x * blockDim.x + threadIdx.x;
  if (t >= (long)Bn * Nu) return;
  const int b = (int)(t / Nu);
  const float px = ux[t * 3 + 0], py = ux[t * 3 + 1], pz = ux[t * 3 + 2];
  float d0 = 1e30f, d1 = 1e30f, d2v = 1e30f;
  int i0 = 0, i1 = 0, i2 = 0;
  const float* kb = kx + (long)b * Nk * 3;
  for (int k = 0; k < Nk; ++k) {
    float dx = kb[k * 3 + 0] - px;
    float dy = kb[k * 3 + 1] - py;
    float dz = kb[k * 3 + 2] - pz;
    float d = dx * dx + dy * dy + dz * dz;
    if (d < d0) {
      d2v = d1; i2 = i1; d1 = d0; i1 = i0; d0 = d; i0 = k;
    } else if (d < d1) {
      d2v = d1; i2 = i1; d1 = d; i1 = k;
    } else if (d < d2v) {
      d2v = d; i2 = k;
    }
  }
  float r0 = 1.0f / (d0 + 1e-8f);
  float r1 = 1.0f / (d1 + 1e-8f);
  float r2 = 1.0f / (d2v + 1e-8f);
  float inv = 1.0f / (r0 + r1 + r2);
  idx3[t * 3 + 0] = i0;
  idx3[t * 3 + 1] = i1;
  idx3[t * 3 + 2] = i2;
  w3[t * 3 + 0] = r0 * inv;
  w3[t * 3 + 1] = r1 * inv;
  w3[t * 3 + 2] = r2 * inv;
}

// out[b][c][n]: c<Ck -> weighted 3-NN interp of kf; else unknown_feats row.
__global__ void interp_concat_kernel(const float* __restrict__ kf,  // B,Ck,Nk
                                     const int* __restrict__ idx3,
                                     const float* __restrict__ w3,
                                     const float* __restrict__ uf,  // B,Cu,Nu
                                     int Bn, int Ck, int Cu, int Nk, int Nu,
                                     float* __restrict__ out) {
  const int C = Ck + Cu;
  long t = (long)blockIdx.x * blockDim.x + threadIdx.x;
  long total = (long)Bn * C * Nu;
  if (t >= total) return;
  const int n = (int)(t % Nu);
  long r = t / Nu;
  const int c = (int)(r % C);
  const int b = (int)(r / C);
  float v;
  if (c < Ck) {
    const long pt = (long)b * Nu + n;
    const int* id = idx3 + pt * 3;
    const float* w = w3 + pt * 3;
    const float* f = kf + ((long)b * Ck + c) * Nk;
    v = w[0] * f[id[0]] + w[1] * f[id[1]] + w[2] * f[id[2]];
  } else {
    v = uf[((long)b * Cu + (c - Ck)) * Nu + n];
  }
  out[t] = v;
}

// ---------------------------------------------------------------------------
// Concat helpers for stage-2 inputs
// ---------------------------------------------------------------------------
__global__ void concat_xyz_kernel(const float* __restrict__ a,   // B,Sa,3
                                  const float* __restrict__ b2,  // B,Sb,3
                                  int Bn, int Sa, int Sb,
                                  float* __restrict__ out) {
  long t = (long)blockIdx.x * blockDim.x + threadIdx.x;
  long total = (long)Bn * (Sa + Sb) * 3;
  if (t >= total) return;
  const int d = (int)(t % 3);
  long r = t / 3;
  const int n = (int)(r % (Sa + Sb));
  const int b = (int)(r / (Sa + Sb));
  out[t] = (n < Sa) ? a[((long)b * Sa + n) * 3 + d]
                    : b2[((long)b * Sb + (n - Sa)) * 3 + d];
}

// fc[b][c][n]: n<S1 -> (c<C1 ? f1 : 0) ; n>=S1 -> (c>=C1 ? rgbf : 0)
__global__ void build_fc_kernel(const float* __restrict__ f1,    // B,C1,S1
                                const float* __restrict__ rgbf,  // B,C1,M
                                int Bn, int C1, int S1, int M,
                                float* __restrict__ out) {
  const int C = 2 * C1, Np = S1 + M;
  long t = (long)blockIdx.x * blockDim.x + threadIdx.x;
  long total = (long)Bn * C * Np;
  if (t >= total) return;
  const int n = (int)(t % Np);
  long r = t / Np;
  const int c = (int)(r % C);
  const int b = (int)(r / C);
  float v = 0.0f;
  if (n < S1) {
    if (c < C1) v = f1[((long)b * C1 + c) * S1 + n];
  } else {
    if (c >= C1) v = rgbf[((long)b * C1 + (c - C1)) * M + (n - S1)];
  }
  out[t] = v;
}

// ---------------------------------------------------------------------------
// Host orchestration
// ---------------------------------------------------------------------------
extern "C" void kernel_launch(void* const* d_in, const int* in_sizes, int n_in,
                              void* d_out, int out_size, void* d_ws,
                              size_t ws_size, hipStream_t stream) {
  (void)in_sizes; (void)n_in; (void)out_size; (void)ws_size;
  const int B = 2, N1 = 8192, S1 = 4096, S2 = 2048, M = 4096;

  const float* pcd_xyz = (const float*)d_in[0];   // B,N1,3
  const float* rgb_xyz = (const float*)d_in[1];   // B,M,3
  const float* rgb_feats = (const float*)d_in[2]; // B,128,M

  // params flat order: sa1 L0..2, sa2 L0..2, fp3 L0..1, fp4 L0..1;
  // each layer = {W, gamma, beta, mean, var}
  auto Wp = [&](int L) { return (const float*)d_in[3 + 5 * L + 0]; };
  auto Gp = [&](int L) { return (const float*)d_in[3 + 5 * L + 1]; };
  auto Bp = [&](int L) { return (const float*)d_in[3 + 5 * L + 2]; };
  auto Mp = [&](int L) { return (const float*)d_in[3 + 5 * L + 3]; };
  auto Vp = [&](int L) { return (const float*)d_in[3 + 5 * L + 4]; };

  const int cout_tab[10] = {64, 64, 128, 256, 256, 256, 256, 256, 256, 256};

  // ---- workspace bump allocator ----
  char* p = (char*)d_ws;
  auto alloc = [&](size_t bytes) -> void* {
    uintptr_t q = ((uintptr_t)p + 255) & ~(uintptr_t)255;
    p = (char*)(q + bytes);
    return (void*)q;
  };
  float *scaleA[10], *biasA[10];
  for (int L = 0; L < 10; ++L) {
    scaleA[L] = (float*)alloc((size_t)cout_tab[L] * 4);
    biasA[L] = (float*)alloc((size_t)cout_tab[L] * 4);
  }
  float* W0pad = (float*)alloc((size_t)64 * 4 * 4);     // 64 x 4
  float* W3pad = (float*)alloc((size_t)256 * 260 * 4);  // 256 x 260
  int* fps1 = (int*)alloc((size_t)B * S1 * 4);
  float* xyz1 = (float*)alloc((size_t)B * S1 * 3 * 4);
  int* idx1 = (int*)alloc((size_t)B * S1 * 64 * 4);
  float* f1 = (float*)alloc((size_t)B * 128 * S1 * 4);
  float* xyzc = (float*)alloc((size_t)B * (S1 + M) * 3 * 4);
  float* fc = (float*)alloc((size_t)B * 256 * (S1 + M) * 4);
  int* fps2 = (int*)alloc((size_t)B * S2 * 4);
  float* xyz2 = (float*)alloc((size_t)B * S2 * 3 * 4);
  int* idx2 = (int*)alloc((size_t)B * S2 * 64 * 4);
  float* f2 = (float*)alloc((size_t)B * 256 * S2 * 4);
  int* nnIdxA = (int*)alloc((size_t)B * S1 * 3 * 4);
  float* nnWA = (float*)alloc((size_t)B * S1 * 3 * 4);
  float* fp3in = (float*)alloc((size_t)B * 384 * S1 * 4);
  float* fp3h = (float*)alloc((size_t)B * 256 * S1 * 4);
  float* fp3o = (float*)alloc((size_t)B * 256 * S1 * 4);
  int* nnIdxB = (int*)alloc((size_t)B * N1 * 3 * 4);
  float* nnWB = (float*)alloc((size_t)B * N1 * 3 * 4);
  float* fp4in = (float*)alloc((size_t)B * 256 * N1 * 4);
  float* fp4h = (float*)alloc((size_t)B * 256 * N1 * 4);

  // ---- BN fusion + weight padding ----
  for (int L = 0; L < 10; ++L)
    bn_fuse_kernel<<<(cout_tab[L] + 255) / 256, 256, 0, stream>>>(
        Gp(L), Bp(L), Mp(L), Vp(L), cout_tab[L], scaleA[L], biasA[L]);
  pad_w_kernel<<<(64 * 4 + 255) / 256, 256, 0, stream>>>(Wp(0), 64, 3, 4,
                                                         W0pad);
  pad_w_kernel<<<(256 * 260 + 255) / 256, 256, 0, stream>>>(Wp(3), 256, 259,
                                                            260, W3pad);

  // ---- SA1 ----
  const size_t fpsShm = (size_t)(4 * N1 + 2 * 1024) * 4;
  hipFuncSetAttribute((const void*)fps_kernel,
                      hipFuncAttributeMaxDynamicSharedMemorySize,
                      (int)(160 * 1024));
  fps_kernel<<<B, 1024, fpsShm, stream>>>(pcd_xyz, N1, S1, fps1);
  gather_xyz_kernel<<<(B * S1 + 255) / 256, 256, 0, stream>>>(pcd_xyz, fps1, B,
                                                              N1, S1, xyz1);
  ball_query_kernel<<<(B * S1 * 32 + 255) / 256, 256, 0, stream>>>(
      pcd_xyz, xyz1, B, N1, S1, 0.01f, idx1);

  hipFuncSetAttribute((const void*)sa_fused_kernel,
                      hipFuncAttributeMaxDynamicSharedMemorySize,
                      (int)(208 * 1024));
  const size_t saShm1 = (size_t)(4 + 2 * 128) * 64 * 4;  // ~66KB
  sa_fused_kernel<<<B * S1, 256, saShm1, stream>>>(
      pcd_xyz, nullptr, xyz1, idx1, W0pad, scaleA[0], biasA[0], 64, Wp(1),
      scaleA[1], biasA[1], 64, Wp(2), scaleA[2], biasA[2], 128, /*Cf=*/0, N1,
      S1, /*1/r=*/10.0f, f1);

  // ---- stage-2 inputs ----
  concat_xyz_kernel<<<((long)B * (S1 + M) * 3 + 255) / 256, 256, 0, stream>>>(
      xyz1, rgb_xyz, B, S1, M, xyzc);
  build_fc_kernel<<<((long)B * 256 * (S1 + M) + 255) / 256, 256, 0, stream>>>(
      f1, rgb_feats, B, 128, S1, M, fc);

  // ---- SA2 ----
  fps_kernel<<<B, 1024, fpsShm, stream>>>(xyzc, S1 + M, S2, fps2);
  gather_xyz_kernel<<<(B * S2 + 255) / 256, 256, 0, stream>>>(xyzc, fps2, B,
                                                              S1 + M, S2, xyz2);
  ball_query_kernel<<<(B * S2 * 32 + 255) / 256, 256, 0, stream>>>(
      xyzc, xyz2, B, S1 + M, S2, 0.04f, idx2);
  const size_t saShm2 = (size_t)(260 + 2 * 256) * 64 * 4;  // ~193KB
  sa_fused_kernel<<<B * S2, 256, saShm2, stream>>>(
      xyzc, fc, xyz2, idx2, W3pad, scaleA[3], biasA[3], 256, Wp(4), scaleA[4],
      biasA[4], 256, Wp(5), scaleA[5], biasA[5], 256, /*Cf=*/256, S1 + M, S2,
      /*1/r=*/5.0f, f2);

  // ---- FP3: xyz1 <- xyz2 ----
  nn3_kernel<<<(B * S1 + 255) / 256, 256, 0, stream>>>(xyz1, xyz2, B, S1, S2,
                                                       nnIdxA, nnWA);
  interp_concat_kernel<<<((long)B * 384 * S1 + 255) / 256, 256, 0, stream>>>(
      f2, nnIdxA, nnWA, f1, B, 256, 128, S2, S1, fp3in);
  {
    long tiles = (long)B * (256 / 16) * (S1 / 16);
    pw_gemm_kernel<<<(int)((tiles + 7) / 8), 256, 0, stream>>>(
        fp3in, Wp(6), scaleA[6], biasA[6], fp3h, B, 384, 256, S1);
    pw_gemm_kernel<<<(int)((tiles + 7) / 8), 256, 0, stream>>>(
        fp3h, Wp(7), scaleA[7], biasA[7], fp3o, B, 256, 256, S1);
  }

  // ---- FP4: pcd_xyz <- xyz1 ----
  nn3_kernel<<<(B * N1 + 255) / 256, 256, 0, stream>>>(pcd_xyz, xyz1, B, N1,
                                                       S1, nnIdxB, nnWB);
  interp_concat_kernel<<<((long)B * 256 * N1 + 255) / 256, 256, 0, stream>>>(
      fp3o, nnIdxB, nnWB, nullptr, B, 256, 0, S1, N1, fp4in);
  {
    long tiles = (long)B * (256 / 16) * (N1 / 16);
    pw_gemm_kernel<<<(int)((tiles + 7) / 8), 256, 0, stream>>>(
        fp4in, Wp(8), scaleA[8], biasA[8], fp4h, B, 256, 256, N1);
    pw_gemm_kernel<<<(int)((tiles + 7) / 8), 256, 0, stream>>>(
        fp4h, Wp(9), scaleA[9], biasA[9], (float*)d_out, B, 256, 256, N1);
  }
}